// neuron_layer_31207232373287
// MI455X (gfx1250) — compile-verified
//
#include <hip/hip_runtime.h>

typedef float v2f __attribute__((ext_vector_type(2)));
typedef float v4f __attribute__((ext_vector_type(4)));
typedef float v8f __attribute__((ext_vector_type(8)));

#define T_STEPS 50
#define BATCH   8192
#define NIN     784
#define NHID    32
#define NOUT    10
#define W1S     788      // padded LDS row stride for W1 (floats); 788%64=20 -> conflict-free b64
#define KC      112      // K staging chunk (112*4B = 448B, 16B-aligned rows)
#define XS      116      // padded LDS row stride for x tile; 116%64=52 -> conflict-free b64
#define WAVES   2        // 1 batch tile per wave -> 256 workgroups cover the whole chip
#define NBUF    3        // triple buffer, prefetch depth 2
#define XTILE   (16 * XS)        // floats per x buffer (1856)
#define XTILEB  (XTILE * 4)      // bytes per x buffer (7424)
#define ALPHA   0.9f
#define BETA    0.9f

__global__ __launch_bounds__(64) void snn_fused_kernel(
    const float* __restrict__ x,   // [T, B, 784]
    const float* __restrict__ W1,  // [32, 784]
    const float* __restrict__ W2,  // [10, 32]
    float* __restrict__ out)       // [B, 10]
{
  extern __shared__ float smem[];
  float* w1lds = smem;                                    // 32 * 788 floats
  const int tid  = threadIdx.x;
  const int lane = tid & 31;
  const int wave = tid >> 5;
  const int lo16 = lane & 15;
  const int hi   = lane >> 4;          // 0 for lanes 0-15, 1 for 16-31
  const bool lo  = (hi == 0);
  float* xt = smem + NHID * W1S + wave * (NBUF * XTILE);  // per-wave triple buffer

  // ---- per-lane staging offsets (14 x b128 per 16x112 chunk) ----
  int goff[14];        // global float offsets within a [16 x 784] row block
  uint32_t lof4[14];   // LDS byte offsets within one padded [16 x 116] buffer
  #pragma unroll
  for (int i = 0; i < 14; ++i) {
    int f = i * 128 + lane * 4;
    int r = f / KC, c = f - r * KC;
    goff[i] = r * NIN + c;
    lof4[i] = (uint32_t)(r * XS + c) * 4u;
  }

  const uint32_t xt_base = (uint32_t)(uintptr_t)xt;       // LDS byte address
  const int b0 = blockIdx.x * (WAVES * 16) + wave * 16;

  // async prefetch of one 16x112 chunk into an LDS buffer (14 x b128 per lane)
  auto prefetch = [&](const float* g, uint32_t lbase) {
    #pragma unroll
    for (int i = 0; i < 14; ++i) {
      uint64_t ga = (uint64_t)(uintptr_t)(g + goff[i]);
      uint32_t la = lbase + lof4[i];
      asm volatile("global_load_async_to_lds_b128 %0, %1, off"
                   :: "v"(la), "v"(ga) : "memory");
    }
  };

  // prime the pipeline with chunks (0,0) and (0,1); overlaps with W1 staging below
  prefetch(x + (size_t)b0 * NIN, xt_base);
  prefetch(x + (size_t)b0 * NIN + KC, xt_base + XTILEB);

  // ---- stage W1 into LDS (padded rows), coalesced b128 ----
  for (int f = tid * 4; f < NHID * NIN; f += 64 * 4) {
    int r = f / NIN, c = f - r * NIN;                     // 784%4==0: no row crossing
    *(v4f*)(w1lds + r * W1S + c) = *(const v4f*)(W1 + f);
  }
  __syncthreads();

  // ---- preload W2^T B-operand into registers (constant across time) ----
  // B[k,n]: lane holds n=lo16 (output o), VGPR j -> k = 4*c2 + 2*hi + j
  v2f w2b[8];
  #pragma unroll
  for (int c2 = 0; c2 < 8; ++c2) {
    if (lo16 < NOUT) {
      w2b[c2] = *(const v2f*)(W2 + lo16 * NHID + 4 * c2 + 2 * hi);
    } else {
      v2f z = {0.f, 0.f};
      w2b[c2] = z;                       // pad N=10..15 with zeros
    }
  }

  // recurrent state, all in WMMA C-layout registers
  v8f syn1[2] = {}, mem1[2] = {};
  v8f syn2 = {}, mem2 = {}, acc = {};

  // A-operand pointers into LDS W1: lane m=lo16, k in VGPRs (k0 + 2*hi + {0,1})
  const float* w1a0 = w1lds + lo16 * W1S + 2 * hi;
  const float* w1a1 = w1a0 + 16 * W1S;
  const float* xb0  = xt + lo16 * XS + 2 * hi;   // B-operand: lane n=lo16 (batch)

  // rotating buffer byte-offsets: o0 = chunk cc (ready), o1 = cc+1, o2 = prefetch target
  int o0 = 0, o1 = XTILEB, o2 = 2 * XTILEB;

  for (int t = 0; t < T_STEPS; ++t) {
    v8f c1a = {}, c1b = {};                       // cur1 accumulators (2 hidden tiles)
    const float* xg = x + ((size_t)t * BATCH + b0) * NIN;

    #pragma unroll
    for (int c = 0; c < 7; ++c) {                 // 7 chunks of K=112
      // prefetch chunk cc+2 into buffer o2, then wait until chunk cc is resident.
      // in-order async completion: <=28 outstanding proves chunk cc's 14 are done.
      if (c <= 4) {
        prefetch(xg + (c + 2) * KC, xt_base + (uint32_t)o2);
        asm volatile("s_wait_asynccnt 0x1c" ::: "memory");
      } else if (c == 5) {
        if (t < T_STEPS - 1) {
          prefetch(xg + (size_t)BATCH * NIN, xt_base + (uint32_t)o2);       // (t+1, 0)
          asm volatile("s_wait_asynccnt 0x1c" ::: "memory");
        } else {
          asm volatile("s_wait_asynccnt 0xe" ::: "memory");
        }
      } else {
        if (t < T_STEPS - 1) {
          prefetch(xg + (size_t)BATCH * NIN + KC, xt_base + (uint32_t)o2);  // (t+1, 1)
          asm volatile("s_wait_asynccnt 0x1c" ::: "memory");
        } else {
          asm volatile("s_wait_asynccnt 0x0" ::: "memory");                 // drain
        }
      }

      const float* xb = (const float*)((const char*)xb0 + o0);
      // 28 K=4 sub-chunks: D[h,b] += W1[h,k] * x[b,k]
      #pragma unroll
      for (int q = 0; q < 28; ++q) {
        int k = c * KC + 4 * q;
        v2f a0 = *(const v2f*)(w1a0 + k);
        v2f a1 = *(const v2f*)(w1a1 + k);
        v2f bx = *(const v2f*)(xb + 4 * q);
        c1a = __builtin_amdgcn_wmma_f32_16x16x4_f32(false, a0, false, bx, (short)0, c1a, false, false);
        c1b = __builtin_amdgcn_wmma_f32_16x16x4_f32(false, a1, false, bx, (short)0, c1b, false, false);
      }
      // rotate buffers
      int tmp = o0; o0 = o1; o1 = o2; o2 = tmp;
    }

    // ---- LIF layer 1 (elementwise in C-layout) ----
    v8f spk[2];
    #pragma unroll
    for (int mt = 0; mt < 2; ++mt) {
      #pragma unroll
      for (int r = 0; r < 8; ++r) {
        float cur = (mt == 0) ? c1a[r] : c1b[r];
        float rst = (mem1[mt][r] > 1.0f) ? 1.0f : 0.0f;   // reset from previous mem
        float s = ALPHA * syn1[mt][r] + cur;
        float m = BETA * mem1[mt][r] + s - rst;
        syn1[mt][r] = s;
        mem1[mt][r] = m;
        spk[mt][r] = (m - 1.0f > 0.0f) ? 1.0f : 0.0f;
      }
    }

    // ---- layer 2: A = spk1 [batch x 32], B = W2^T (in regs) ----
    // D1 lane pair (l, l+16) shares batch b0+lo16; lo holds h=16mt+r, hi holds h=16mt+8+r.
    // A chunk c2 needs: lo -> h=k0+{0,1}, hi -> h=k0+2+{0,1}; half come from the partner lane.
    v8f cur2 = {};
    #pragma unroll
    for (int c2 = 0; c2 < 8; ++c2) {
      int mt = c2 >> 2, q = c2 & 3;
      int base = (q & 1) * 4;
      float a0, a1;
      if (q < 2) {          // k0 mod 16 in {0,4}: lo owns, hi fetches from lo
        float w0  = __shfl_xor(spk[mt][base + 2], 16, 32);
        float w1v = __shfl_xor(spk[mt][base + 3], 16, 32);
        a0 = lo ? spk[mt][base + 0] : w0;
        a1 = lo ? spk[mt][base + 1] : w1v;
      } else {              // k0 mod 16 in {8,12}: hi owns, lo fetches from hi
        float w0  = __shfl_xor(spk[mt][base + 0], 16, 32);
        float w1v = __shfl_xor(spk[mt][base + 1], 16, 32);
        a0 = lo ? w0  : spk[mt][base + 2];
        a1 = lo ? w1v : spk[mt][base + 3];
      }
      v2f a; a[0] = a0; a[1] = a1;
      cur2 = __builtin_amdgcn_wmma_f32_16x16x4_f32(false, a, false, w2b[c2], (short)0, cur2, false, false);
    }

    // ---- LIF layer 2 + spike accumulation (padded o>=10 stays exactly 0) ----
    #pragma unroll
    for (int r = 0; r < 8; ++r) {
      float rst = (mem2[r] > 1.0f) ? 1.0f : 0.0f;
      float s = ALPHA * syn2[r] + cur2[r];
      float m = BETA * mem2[r] + s - rst;
      syn2[r] = s;
      mem2[r] = m;
      acc[r] += (m - 1.0f > 0.0f) ? 1.0f : 0.0f;
    }
  }

  // ---- normalize by row sum and store ----
  // D2 layout: VGPR r -> batch b0 + r + 8*hi; lane -> o = lo16
  #pragma unroll
  for (int r = 0; r < 8; ++r) {
    float s = acc[r];
    s += __shfl_xor(s, 1, 32);
    s += __shfl_xor(s, 2, 32);
    s += __shfl_xor(s, 4, 32);
    s += __shfl_xor(s, 8, 32);       // sum over the 16-lane group (o=10..15 are zero)
    float safe = (s == 0.0f) ? 1.0f : s;
    if (lo16 < NOUT) {
      int b = b0 + r + 8 * hi;
      out[(size_t)b * NOUT + lo16] = acc[r] / safe;
    }
  }
}

extern "C" void kernel_launch(void* const* d_in, const int* in_sizes, int n_in,
                              void* d_out, int out_size, void* d_ws, size_t ws_size,
                              hipStream_t stream) {
  const float* x  = (const float*)d_in[0];
  const float* W1 = (const float*)d_in[1];
  const float* W2 = (const float*)d_in[2];
  float* out = (float*)d_out;

  const size_t shmem = (size_t)(NHID * W1S + WAVES * NBUF * XTILE) * sizeof(float); // ~145 KB
  hipFuncSetAttribute((const void*)snn_fused_kernel,
                      hipFuncAttributeMaxDynamicSharedMemorySize, (int)shmem);
  dim3 grid(BATCH / (WAVES * 16));   // 256 workgroups
  dim3 block(64);                    // 2 waves, 1 batch tile each
  hipLaunchKernelGGL(snn_fused_kernel, grid, block, shmem, stream, x, W1, W2, out);
}